// RLDAConv_55903294325354
// MI455X (gfx1250) — compile-verified
//
#include <hip/hip_runtime.h>

#define BATCH 8
#define CIN 64
#define HWD 56
#define LPB (HWD * HWD)          // 3136 patches per image
#define NPATCH (BATCH * LPB)     // 25088
#define DDIM 576                 // 64 * 3 * 3
#define KCENT 256
#define COUT 128
#define WG 64                    // patches per workgroup in search kernels

typedef __attribute__((ext_vector_type(2))) float v2f;
typedef __attribute__((ext_vector_type(8))) float v8f;

// ---------------------------------------------------------------------------
// Async-stage one centroid row (576 f32 = 2304 B = 144 x b128) from global
// into LDS using the CDNA5 async copy path (tracked by ASYNCcnt).
// Transfers split across the 64 threads: t = tid (offset 0), t = tid + 64
// (offset:1024), t = tid + 128 for tid < 16 (offset:2048). The 24-bit
// instruction offset applies to both the global and the LDS address.
// ---------------------------------------------------------------------------
__device__ __forceinline__ void stage_centroid(const float* __restrict__ cent_row,
                                               float* dst_lds, int tid) {
    const unsigned int lds = (unsigned int)(unsigned long long)dst_lds +
                             (unsigned int)tid * 16u;
    const unsigned int go = (unsigned int)tid * 16u;
    asm volatile(
        "global_load_async_to_lds_b128 %0, %1, %2\n\t"
        "global_load_async_to_lds_b128 %0, %1, %2 offset:1024"
        :: "v"(lds), "v"(go), "s"(cent_row) : "memory");
    if (tid < 16) {
        asm volatile(
            "global_load_async_to_lds_b128 %0, %1, %2 offset:2048"
            :: "v"(lds), "v"(go), "s"(cent_row) : "memory");
    }
}

__device__ __forceinline__ void wait_async_all() {
    asm volatile("s_wait_asynccnt 0x0" ::: "memory");
}

// ---------------------------------------------------------------------------
// L1 nearest-centroid search with double-buffered async centroid staging.
// Patch data lives in LDS, vec4-interleaved: fl[(d/4)*(WG*4) + tid*4 + d%4]
// -> each lane issues ds_load_b128. Centroid reads are same-address LDS
// broadcasts (conflict-free). Copy of k+1 overlaps compute of k.
// ---------------------------------------------------------------------------
__device__ __forceinline__ int search_staged(const float* __restrict__ cent,
                                             const float* fl,
                                             float* cb0, float* cb1, int tid) {
    stage_centroid(cent, cb0, tid);
    wait_async_all();
    __syncthreads();

    float best = 3.4e38f;
    int bi = 0;
    for (int k = 0; k < KCENT; ++k) {
        if (k + 1 < KCENT)
            stage_centroid(cent + (size_t)(k + 1) * DDIM,
                           (k & 1) ? cb0 : cb1, tid);
        const float* ck = (k & 1) ? cb1 : cb0;
        float acc = 0.f;
#pragma unroll 4
        for (int dv = 0; dv < DDIM / 4; ++dv) {
            const float4 c4 = *(const float4*)(ck + dv * 4);
            const float4 f4 = *(const float4*)(fl + dv * (WG * 4) + tid * 4);
            acc += fabsf(f4.x - c4.x) + fabsf(f4.y - c4.y) +
                   fabsf(f4.z - c4.z) + fabsf(f4.w - c4.w);
        }
        if (acc < best) { best = acc; bi = k; }  // strict < : first-min ties
        wait_async_all();     // own wave's k+1 transfers landed
        __syncthreads();      // all waves done reading buf k, buf k+1 visible
    }
    return bi;
}

// ---------------------------------------------------------------------------
// Stage 1: im2col one patch per thread into LDS, then L1 argmin vs centroids.
// ---------------------------------------------------------------------------
__global__ __launch_bounds__(WG) void k_stage1(const float* __restrict__ x,
                                               const float* __restrict__ cent,
                                               int* __restrict__ idx1) {
    __shared__ float fl[WG * DDIM];      // 147456 B
    __shared__ float cbuf[2][DDIM];      // 4608 B double buffer
    const int tid = threadIdx.x;
    const int p = blockIdx.x * WG + tid;
    const int b = p / LPB, l = p % LPB;
    const int oy = l / HWD, ox = l % HWD;
    const float* xb = x + (size_t)b * CIN * LPB;
    for (int c = 0; c < CIN; ++c) {
        const float* xc = xb + (size_t)c * LPB;
#pragma unroll
        for (int i = 0; i < 3; ++i) {
            const int y = oy + i - 1;
#pragma unroll
            for (int j = 0; j < 3; ++j) {
                const int xx = ox + j - 1;
                float v = 0.f;
                if (y >= 0 && y < HWD && xx >= 0 && xx < HWD) v = xc[y * HWD + xx];
                const int d = c * 9 + i * 3 + j;
                fl[(d >> 2) * (WG * 4) + tid * 4 + (d & 3)] = v;
            }
        }
    }
    // each thread reads only its own patch slots; cbuf synced in search_staged
    idx1[p] = search_staged(cent, fl, cbuf[0], cbuf[1], tid);
}

// ---------------------------------------------------------------------------
// Stage 2: residual = patch - centroids[idx1], then L1 argmin vs res-centroids.
// ---------------------------------------------------------------------------
__global__ __launch_bounds__(WG) void k_stage2(const float* __restrict__ x,
                                               const float* __restrict__ cent,
                                               const float* __restrict__ rcent,
                                               const int* __restrict__ idx1,
                                               int* __restrict__ idx2) {
    __shared__ float fl[WG * DDIM];
    __shared__ float cbuf[2][DDIM];
    const int tid = threadIdx.x;
    const int p = blockIdx.x * WG + tid;
    const int b = p / LPB, l = p % LPB;
    const int oy = l / HWD, ox = l % HWD;
    const float* xb = x + (size_t)b * CIN * LPB;
    const float* c1 = cent + (size_t)idx1[p] * DDIM;
    for (int c = 0; c < CIN; ++c) {
        const float* xc = xb + (size_t)c * LPB;
#pragma unroll
        for (int i = 0; i < 3; ++i) {
            const int y = oy + i - 1;
#pragma unroll
            for (int j = 0; j < 3; ++j) {
                const int xx = ox + j - 1;
                float v = 0.f;
                if (y >= 0 && y < HWD && xx >= 0 && xx < HWD) v = xc[y * HWD + xx];
                const int d = c * 9 + i * 3 + j;
                fl[(d >> 2) * (WG * 4) + tid * 4 + (d & 3)] = v - c1[d];
            }
        }
    }
    idx2[p] = search_staged(rcent, fl, cbuf[0], cbuf[1], tid);
}

// ---------------------------------------------------------------------------
// Histograms: h1[k], h2[k] and joint H12[k1][k2] (f32, global atomic adds).
// ---------------------------------------------------------------------------
__global__ void k_hist(const int* __restrict__ i1, const int* __restrict__ i2,
                       float* __restrict__ h1, float* __restrict__ h2,
                       float* __restrict__ H12) {
    const int p = blockIdx.x * blockDim.x + threadIdx.x;
    if (p >= NPATCH) return;
    const int a = i1[p], b = i2[p];
    atomicAdd(&h1[a], 1.f);
    atomicAdd(&h2[b], 1.f);
    atomicAdd(&H12[a * KCENT + b], 1.f);
}

// ---------------------------------------------------------------------------
// WMMA: T = H12 (256x256) @ dotrc (256x128), exact f32 via
// V_WMMA_F32_16X16X4_F32. One wave per 16x16 tile of T; K accumulated 4 at a
// time. A 16x4 operand: 2 VGPRs, lanes 0-15 hold K={0,1}, lanes 16-31 K={2,3}.
// B 4x16 operand: symmetric. C/D: 8 VGPRs, lanes 16-31 carry M+8.
// ---------------------------------------------------------------------------
__global__ __launch_bounds__(32) void k_wmma_T(const float* __restrict__ H12,
                                               const float* __restrict__ dotrc,
                                               float* __restrict__ T) {
    const int tile = blockIdx.x;        // 0 .. (16*8 - 1)
    const int nt = tile & 7;            // COUT/16 = 8 column tiles
    const int mt = tile >> 3;           // KCENT/16 = 16 row tiles
    const int lane = threadIdx.x;
    const int half = lane >> 4;         // 0: lanes 0-15, 1: lanes 16-31
    const int lm = lane & 15;
    v8f acc = {};
    for (int kk = 0; kk < KCENT / 4; ++kk) {
        const float* arow = H12 + (size_t)(mt * 16 + lm) * KCENT + kk * 4 + half * 2;
        v2f a;
        a.x = arow[0];                  // K = half*2
        a.y = arow[1];                  // K = half*2 + 1
        const float* bcol = dotrc + (size_t)(kk * 4 + half * 2) * COUT + nt * 16 + lm;
        v2f bb;
        bb.x = bcol[0];                 // row K = half*2
        bb.y = bcol[COUT];              // row K = half*2 + 1
        acc = __builtin_amdgcn_wmma_f32_16x16x4_f32(false, a, false, bb,
                                                    (short)0, acc, false, false);
    }
#pragma unroll
    for (int r = 0; r < 8; ++r) {
        const int row = mt * 16 + r + half * 8;
        const int col = nt * 16 + lm;
        T[(size_t)row * COUT + col] = acc[r];
    }
}

// ---------------------------------------------------------------------------
// Per-channel BN moments from histograms + WMMA cross term.
// mean = (h1.dotc + h2.dotrc)/N ;  E[x^2] uses 2*sum_k dotc[k,c]*T[k,c].
// ---------------------------------------------------------------------------
__global__ __launch_bounds__(COUT) void k_stats(const float* __restrict__ h1,
                                                const float* __restrict__ h2,
                                                const float* __restrict__ dotc,
                                                const float* __restrict__ dotrc,
                                                const float* __restrict__ T,
                                                float* __restrict__ meanv,
                                                float* __restrict__ invstd) {
    const int co = threadIdx.x;
    float s = 0.f, sq = 0.f, cross = 0.f;
    for (int k = 0; k < KCENT; ++k) {
        const float dc = dotc[k * COUT + co];
        const float dr = dotrc[k * COUT + co];
        const float a = h1[k], b = h2[k];
        s += a * dc + b * dr;
        sq += a * dc * dc + b * dr * dr;
        cross += dc * T[k * COUT + co];
    }
    sq += 2.f * cross;
    const float n = (float)NPATCH;
    const float m = s / n;
    const float var = sq / n - m * m;
    meanv[co] = m;
    invstd[co] = rsqrtf(var + 1e-5f);
}

// ---------------------------------------------------------------------------
// Final output: gather LUTs, normalize, write [B, Cout, oH, oW].
// ---------------------------------------------------------------------------
__global__ void k_out(const int* __restrict__ i1, const int* __restrict__ i2,
                      const float* __restrict__ dotc, const float* __restrict__ dotrc,
                      const float* __restrict__ meanv, const float* __restrict__ invstd,
                      const float* __restrict__ gamma, const float* __restrict__ beta,
                      float* __restrict__ out) {
    const int e = blockIdx.x * blockDim.x + threadIdx.x;
    if (e >= NPATCH * COUT) return;
    const int b = e / (COUT * LPB);
    const int rem = e % (COUT * LPB);
    const int co = rem / LPB;
    const int l = rem % LPB;
    const int p = b * LPB + l;
    const float v = dotc[i1[p] * COUT + co] + dotrc[i2[p] * COUT + co];
    out[e] = (v - meanv[co]) * invstd[co] * gamma[co] + beta[co];
}

// ---------------------------------------------------------------------------
extern "C" void kernel_launch(void* const* d_in, const int* in_sizes, int n_in,
                              void* d_out, int out_size, void* d_ws, size_t ws_size,
                              hipStream_t stream) {
    (void)in_sizes; (void)n_in; (void)out_size; (void)ws_size;
    const float* x     = (const float*)d_in[0];
    const float* cent  = (const float*)d_in[1];
    const float* rcent = (const float*)d_in[2];
    const float* dotc  = (const float*)d_in[3];
    const float* dotrc = (const float*)d_in[4];
    const float* gamma = (const float*)d_in[5];
    const float* beta  = (const float*)d_in[6];

    char* ws = (char*)d_ws;
    int*   idx1 = (int*)(ws + 0);                 // 25088 * 4
    int*   idx2 = (int*)(ws + 100352);            // 25088 * 4
    float* h1   = (float*)(ws + 200704);          // 256 * 4
    float* h2   = (float*)(ws + 201728);          // 256 * 4
    float* H12  = (float*)(ws + 202752);          // 65536 * 4
    float* T    = (float*)(ws + 464896);          // 32768 * 4
    float* mn   = (float*)(ws + 595968);          // 128 * 4
    float* isd  = (float*)(ws + 596480);          // 128 * 4

    // zero h1 + h2 + H12 (contiguous) every call: deterministic accumulation
    (void)hipMemsetAsync(ws + 200704, 0, (256 + 256 + 65536) * sizeof(float), stream);

    k_stage1<<<NPATCH / WG, WG, 0, stream>>>(x, cent, idx1);
    k_stage2<<<NPATCH / WG, WG, 0, stream>>>(x, cent, rcent, idx1, idx2);
    k_hist<<<(NPATCH + 255) / 256, 256, 0, stream>>>(idx1, idx2, h1, h2, H12);
    k_wmma_T<<<(KCENT / 16) * (COUT / 16), 32, 0, stream>>>(H12, dotrc, T);
    k_stats<<<1, COUT, 0, stream>>>(h1, h2, dotc, dotrc, T, mn, isd);
    k_out<<<(NPATCH * COUT) / 256, 256, 0, stream>>>(idx1, idx2, dotc, dotrc,
                                                     mn, isd, gamma, beta,
                                                     (float*)d_out);
}